// SimGCL_Encoder_6708738916589
// MI455X (gfx1250) — compile-verified
//
#include <hip/hip_runtime.h>
#include <hip/hip_bf16.h>
#include <math.h>

#define USER_NUM 50000
#define ITEM_NUM 50000
#define N_NODES  (USER_NUM + ITEM_NUM)
#define N_EDGES  3200000
#define EMB      64
#define N_LAYERS 3
#define EPS_F    0.1f

typedef __attribute__((ext_vector_type(2))) float v2f;
typedef __attribute__((ext_vector_type(8))) float v8f;

// ---------------------------------------------------------------------------
// Zero a float buffer (float4 granularity; n4 = count of float4)
// ---------------------------------------------------------------------------
__global__ void k_zero(float4* __restrict__ p, int n4) {
    int i = blockIdx.x * blockDim.x + threadIdx.x;
    if (i < n4) {
        float4 z; z.x = 0.f; z.y = 0.f; z.z = 0.f; z.w = 0.f;
        p[i] = z;
    }
}

// ---------------------------------------------------------------------------
// dst = concat(user_emb, item_emb)   (float4 granularity)
// ---------------------------------------------------------------------------
__global__ void k_concat(const float4* __restrict__ u,
                         const float4* __restrict__ it,
                         float4* __restrict__ dst) {
    const int U4 = USER_NUM * EMB / 4;
    const int T4 = N_NODES * EMB / 4;
    int i = blockIdx.x * blockDim.x + threadIdx.x;
    if (i >= T4) return;
    dst[i] = (i < U4) ? u[i] : it[i - U4];
}

// ---------------------------------------------------------------------------
// COO SpMM scatter: y[rows[e]] += vals[e] * x[cols[e]]
// 16 lanes per edge, one float4 (4 dims of 64) per lane.
// Gathers/scatters stay L2-resident (x and y are 25.6MB each, L2 = 192MB).
// unsafeAtomicAdd -> native global_atomic_add_f32 (no CAS loop).
// ---------------------------------------------------------------------------
__global__ void k_spmm(const float* __restrict__ x,
                       const int*   __restrict__ rows,
                       const int*   __restrict__ cols,
                       const float* __restrict__ vals,
                       float*       __restrict__ y) {
    int t    = blockIdx.x * blockDim.x + threadIdx.x;
    int edge = t >> 4;
    int sub  = t & 15;
    if (edge >= N_EDGES) return;

    int   r = rows[edge];
    int   c = cols[edge];
    float v = vals[edge];

    const float4 xv = *(const float4*)(x + (size_t)c * EMB + sub * 4);
    float* yp = y + (size_t)r * EMB + sub * 4;
    unsafeAtomicAdd(yp + 0, v * xv.x);
    unsafeAtomicAdd(yp + 1, v * xv.y);
    unsafeAtomicAdd(yp + 2, v * xv.z);
    unsafeAtomicAdd(yp + 3, v * xv.w);
}

// ---------------------------------------------------------------------------
// Noise step: e += sign(e) * (nz / max(||nz||,1e-12)) * EPS ;  sum += e
// One wave handles 16 nodes. Row norms ||nz||^2 are computed with
// V_WMMA_F32_16X16X4_F32: A = squared noise chunks (16x4), B = ones (4x16),
// C accumulates over 16 k-chunks -> D[m][n] = ||nz_m||^2 for every n.
// C/D layout: lane n<16 holds M=0..7 in c[0..7]; lane 16+n holds M=8..15.
// Update mapping: lane -> node = base + (lane>=16)*8 + (lane&15)/2,
// covering half a row (32 dims) each; that node's norm is exactly c[(lane&15)/2].
// ---------------------------------------------------------------------------
__global__ void k_noise(float*       __restrict__ e,
                        const float* __restrict__ nz,
                        float*       __restrict__ sum) {
    const int lane = threadIdx.x & 31;
    const int wave = blockIdx.x * (blockDim.x >> 5) + (threadIdx.x >> 5);
    const int NWAVES = N_NODES / 16;          // 6250, exact
    if (wave >= NWAVES) return;               // wave-uniform: EXEC stays all-1s
    const int nodeBase = wave * 16;

    // ---- norms of 16 noise rows via WMMA ----
    const int   row  = lane & 15;
    const int   koff = (lane < 16) ? 0 : 2;   // A 16x4 f32 layout
    const float* nrow = nz + (size_t)(nodeBase + row) * EMB + koff;

    v8f c = {};
#if defined(__has_builtin) && __has_builtin(__builtin_amdgcn_wmma_f32_16x16x4_f32)
    v2f b; b.x = 1.0f; b.y = 1.0f;
#pragma unroll
    for (int kc = 0; kc < 16; ++kc) {
        float x0 = nrow[kc * 4];
        float x1 = nrow[kc * 4 + 1];
        v2f a; a.x = x0 * x0; a.y = x1 * x1;
        c = __builtin_amdgcn_wmma_f32_16x16x4_f32(
                false, a, false, b, (short)0, c, false, false);
    }
#else
    // VALU fallback: partial sums per lane, combined below via the same layout
    // (emulate D[m][*] = norm^2 of row m): reduce across the 4 K slots.
    float part = 0.f;
#pragma unroll
    for (int kc = 0; kc < 16; ++kc) {
        float x0 = nrow[kc * 4];
        float x1 = nrow[kc * 4 + 1];
        part += x0 * x0 + x1 * x1;
    }
    // lane m and lane m+16 hold the two halves of row m's norm
    part += __shfl_xor(part, 16, 32);
    // scatter into c[] mimicking the WMMA C/D layout
#pragma unroll
    for (int m = 0; m < 8; ++m) {
        float lo = __shfl(part, m, 32);       // rows 0..7
        float hi = __shfl(part, m + 8, 32);   // rows 8..15
        c[m] = (lane < 16) ? lo : hi;
    }
#endif

    // ---- per-lane scale for its node ----
    const int j = (lane & 15) >> 1;           // 0..7
    float n2 = (j & 4) ? ((j & 2) ? ((j & 1) ? c[7] : c[6])
                                  : ((j & 1) ? c[5] : c[4]))
                       : ((j & 2) ? ((j & 1) ? c[3] : c[2])
                                  : ((j & 1) ? c[1] : c[0]));
    const float scale = EPS_F / fmaxf(__builtin_sqrtf(n2), 1e-12f);

    const int node = nodeBase + ((lane >> 4) << 3) + j;
    const int half = lane & 1;                // which 32-dim half of the row
    const size_t base = (size_t)node * EMB + half * 32;

#pragma unroll
    for (int d = 0; d < 32; d += 4) {
        float4 ev = *(const float4*)(e   + base + d);
        float4 nv = *(const float4*)(nz  + base + d);
        float4 sv = *(const float4*)(sum + base + d);

        float sx = (ev.x > 0.f) ? 1.f : ((ev.x < 0.f) ? -1.f : 0.f);
        float sy = (ev.y > 0.f) ? 1.f : ((ev.y < 0.f) ? -1.f : 0.f);
        float sz = (ev.z > 0.f) ? 1.f : ((ev.z < 0.f) ? -1.f : 0.f);
        float sw = (ev.w > 0.f) ? 1.f : ((ev.w < 0.f) ? -1.f : 0.f);

        ev.x += sx * nv.x * scale;
        ev.y += sy * nv.y * scale;
        ev.z += sz * nv.z * scale;
        ev.w += sw * nv.w * scale;

        sv.x += ev.x; sv.y += ev.y; sv.z += ev.z; sv.w += ev.w;

        *(float4*)(e   + base + d) = ev;
        *(float4*)(sum + base + d) = sv;
    }
}

// ---------------------------------------------------------------------------
// out *= 1/3 (mean over layers), in place
// ---------------------------------------------------------------------------
__global__ void k_final(float4* __restrict__ out, int n4) {
    int i = blockIdx.x * blockDim.x + threadIdx.x;
    if (i < n4) {
        float4 v = out[i];
        const float s = 1.0f / 3.0f;
        v.x *= s; v.y *= s; v.z *= s; v.w *= s;
        out[i] = v;
    }
}

// ---------------------------------------------------------------------------
extern "C" void kernel_launch(void* const* d_in, const int* in_sizes, int n_in,
                              void* d_out, int out_size, void* d_ws, size_t ws_size,
                              hipStream_t stream) {
    const float* user  = (const float*)d_in[0];
    const float* item  = (const float*)d_in[1];
    const float* vals  = (const float*)d_in[2];
    const float* noise = (const float*)d_in[3];
    const int*   rows  = (const int*)d_in[4];
    const int*   cols  = (const int*)d_in[5];
    float*       out   = (float*)d_out;          // doubles as the layer-sum buffer

    float* bufA = (float*)d_ws;                  // current embeddings
    float* bufB = bufA + (size_t)N_NODES * EMB;  // next embeddings

    const int TOT   = N_NODES * EMB;             // 6.4M floats
    const int TOT4  = TOT / 4;                   // 1.6M float4
    const int ZBLK  = (TOT4 + 255) / 256;        // 6250
    const int SBLK  = (N_EDGES * 16) / 256;      // 200000
    const int NWAVES = N_NODES / 16;             // 6250
    const int NBLK  = (NWAVES + 7) / 8;          // 782 blocks of 8 waves

    // sum accumulator (= d_out) must start at zero every call
    k_zero<<<ZBLK, 256, 0, stream>>>((float4*)out, TOT4);
    // ego = concat(user, item)
    k_concat<<<ZBLK, 256, 0, stream>>>((const float4*)user, (const float4*)item,
                                       (float4*)bufA);

    for (int l = 0; l < N_LAYERS; ++l) {
        k_zero<<<ZBLK, 256, 0, stream>>>((float4*)bufB, TOT4);
        k_spmm<<<SBLK, 256, 0, stream>>>(bufA, rows, cols, vals, bufB);
        k_noise<<<NBLK, 256, 0, stream>>>(bufB,
                                          noise + (size_t)l * TOT,
                                          out);
        float* t = bufA; bufA = bufB; bufB = t;  // next layer reads this output
    }

    k_final<<<ZBLK, 256, 0, stream>>>((float4*)out, TOT4);
}